// LinearCaps_16320875725237
// MI455X (gfx1250) — compile-verified
//
#include <hip/hip_runtime.h>

// Problem constants (from reference)
#define DBIG 1152   // in-capsule count D
#define BB   64     // batch
#define CIN  8      // in-channels C
#define DOUT 64     // out-capsule count d
#define COUT 16     // out-channels c

typedef float v2f __attribute__((ext_vector_type(2)));
typedef float v4f __attribute__((ext_vector_type(4)));
typedef float v8f __attribute__((ext_vector_type(8)));

// Workspace layout (in floats):
//   bl  : routing logits b        [BB][DOUT][DBIG]
//   lse : log-sum-exp over d      [BB][DBIG]
//   s   : capsule pre-activations [BB][DOUT][COUT]
//   v   : squash(s)               [BB][DOUT][COUT]
static constexpr size_t OFF_BL  = 0;
static constexpr size_t OFF_LSE = OFF_BL  + (size_t)BB * DOUT * DBIG;
static constexpr size_t OFF_S   = OFF_LSE + (size_t)BB * DBIG;
static constexpr size_t OFF_V   = OFF_S   + (size_t)BB * DOUT * COUT;

// WMMA f32 16x16x4 layouts (wave32):
//   A (16x4, M=batch, K=C): lanes 0-15 -> M=lane,    VGPR{0,1}=K{0,1};
//                           lanes 16-31 -> M=lane-16, VGPR{0,1}=K{2,3}
//   B (4x16, K=C, N=c):     lane&15 = N; lo half K{0,1}, hi half K{2,3}
//   C/D (16x16):            VGPR r: lanes 0-15 -> M=r, N=lane;
//                                   lanes 16-31 -> M=r+8, N=lane-16

// Two chained K=4 WMMAs computing u_tile(16b x 16c) = X(16x8) @ W^T(8x16),
// with an optional pre-scale of the A rows.
__device__ inline v8f u_tile_wmma(v2f a_lo, v2f a_hi, const float* __restrict__ wr,
                                  int kb, v8f acc) {
  v2f b_lo = *(const v2f*)(wr + kb);
  v2f b_hi = *(const v2f*)(wr + 4 + kb);
  acc = __builtin_amdgcn_wmma_f32_16x16x4_f32(false, a_lo, false, b_lo,
                                              (short)0, acc, false, false);
  acc = __builtin_amdgcn_wmma_f32_16x16x4_f32(false, a_hi, false, b_hi,
                                              (short)0, acc, false, false);
  return acc;
}

// ---------------------------------------------------------------------------
// b <- 0
__global__ void caps_zero_kernel(float* __restrict__ p, size_t n) {
  size_t i = (size_t)blockIdx.x * blockDim.x + threadIdx.x;
  if (i < n) p[i] = 0.0f;
}

// lse[b][Dp] = max_d bl + log(sum_d exp(bl - max))   (softmax over d axis)
__global__ void caps_lse_kernel(const float* __restrict__ bl,
                                float* __restrict__ lse) {
  const int tid = blockIdx.x * blockDim.x + threadIdx.x;
  if (tid >= BB * DBIG) return;
  const int b  = tid / DBIG;
  const float* p = bl + (size_t)b * DOUT * DBIG + (tid % DBIG);
  float mx = -1e30f;
  for (int d = 0; d < DOUT; ++d) mx = fmaxf(mx, p[(size_t)d * DBIG]);
  float z = 0.0f;
  for (int d = 0; d < DOUT; ++d) z += __expf(p[(size_t)d * DBIG] - mx);
  lse[tid] = mx + __logf(z);
}

// s[b,d,c] = sum_D cpl[b,d,D] * u[b,d,D,c],  v = squash(s) = s/(1+|s|^2).
// cpl scales rows of u -> folded into the WMMA A operand; the whole sum over
// D lives in the chained WMMA C accumulator. One WG per (batch-tile, d);
// 8 waves stride D by 8; LDS cross-wave reduction; squash via xor-shuffles.
__global__ __launch_bounds__(256) void caps_s_kernel(
    const float* __restrict__ x, const float* __restrict__ w,
    const float* __restrict__ bl, const float* __restrict__ lse,
    float* __restrict__ s_out, float* __restrict__ v_out) {
  __shared__ float red[8 * 256];
  const int d    = blockIdx.x;
  const int b0   = blockIdx.y * 16;
  const int t    = threadIdx.x;
  const int wave = t >> 5;
  const int lane = t & 31;
  const int hi   = lane >> 4;
  const int m    = lane & 15;          // A row (batch) / B,D column (c)
  const int kb   = hi << 1;            // K pair: {0,1} or {2,3}

  const float* xr  = x + ((size_t)(b0 + m) * CIN) * DBIG;       // + ci*DBIG + Dp
  const float* blr = bl + ((size_t)(b0 + m) * DOUT + d) * DBIG; // + Dp
  const float* lsr = lse + (size_t)(b0 + m) * DBIG;             // + Dp
  const float* wb  = w + (size_t)d * DBIG * COUT * CIN + (size_t)m * CIN;

  v8f acc = {};

  for (int Dp = wave; Dp < DBIG; Dp += 8) {
    const float* wr = wb + (size_t)Dp * (COUT * CIN);
    __builtin_prefetch(wr + 8 * COUT * CIN, 0, 1);  // next iteration's row

    float e = __expf(blr[Dp] - lsr[Dp]);            // coupling coefficient

    v2f a_lo, a_hi;                                 // A = cpl * x
    a_lo.x = e * xr[(size_t)(kb + 0) * DBIG + Dp];
    a_lo.y = e * xr[(size_t)(kb + 1) * DBIG + Dp];
    a_hi.x = e * xr[(size_t)(kb + 4) * DBIG + Dp];
    a_hi.y = e * xr[(size_t)(kb + 5) * DBIG + Dp];

    acc = u_tile_wmma(a_lo, a_hi, wr, kb, acc);
  }

#pragma unroll
  for (int r = 0; r < 8; ++r) red[wave * 256 + r * 32 + lane] = acc[r];
  __syncthreads();

  if (wave == 0) {
#pragma unroll
    for (int r = 0; r < 8; ++r) {
      float sv = 0.0f;
      for (int wv = 0; wv < 8; ++wv) sv += red[wv * 256 + r * 32 + lane];
      // squash: |s|^2 over c = reduce across the 16 lanes of the half-wave
      float sq = sv * sv;
#pragma unroll
      for (int msk = 1; msk < 16; msk <<= 1) sq += __shfl_xor(sq, msk, 32);
      float vv = sv / (1.0f + sq);
      const int bat = b0 + hi * 8 + r;
      const size_t idx = ((size_t)bat * DOUT + d) * COUT + m;
      s_out[idx] = sv;
      v_out[idx] = vv;
    }
  }
}

// bl[b,d,D] += sum_c u[b,d,D,c] * v[b,d,c]
// u recomputed with chained WMMAs. The row-wise dot over c is done via a
// wave-private LDS transpose: each wave computes TWO u tiles (Dp, Dp+1),
// stores them to its own 2KB LDS slot, then each lane owns one
// (batch-row, Dp) output: 4x ds_load_b128 + 16 FMAs + one exclusive RMW.
// Fully convergent (EXEC all-ones), no cross-wave sync needed.
__global__ __launch_bounds__(256) void caps_b_kernel(
    const float* __restrict__ x, const float* __restrict__ w,
    const float* __restrict__ v_in, float* __restrict__ bl) {
  __shared__ float tbuf[8 * 512];      // 2 tiles of 256 floats per wave
  const int d    = blockIdx.x;
  const int b0   = blockIdx.y * 16;
  const int t    = threadIdx.x;
  const int wave = t >> 5;
  const int lane = t & 31;
  const int hi   = lane >> 4;
  const int m    = lane & 15;
  const int kb   = hi << 1;

  const float* xr = x + ((size_t)(b0 + m) * CIN) * DBIG;
  const float* wb = w + (size_t)d * DBIG * COUT * CIN + (size_t)m * CIN;
  float* tb = tbuf + wave * 512;

  // Consumer-side identities: lane owns (batch row = m, tile T = hi)
  const int half = m >> 3;             // which 16-lane half held this row
  const int rrow = m & 7;              // accumulator VGPR index of this row
  const float* tread = tb + hi * 256 + rrow * 32 + half * 16;   // 16 floats
  float* blrow = bl + ((size_t)(b0 + m) * DOUT + d) * DBIG;     // + Dp

  // v is D-invariant: lane hoists the 16 values v[b0+m, d, 0..15]
  v4f vq[4];
  {
    const v4f* vp = (const v4f*)(v_in + ((size_t)(b0 + m) * DOUT + d) * COUT);
#pragma unroll
    for (int q = 0; q < 4; ++q) vq[q] = vp[q];
  }

  for (int Dp = wave * 2; Dp < DBIG; Dp += 16) {   // pair (Dp, Dp+1) per wave
    const float* wr0 = wb + (size_t)Dp * (COUT * CIN);
    const float* wr1 = wr0 + (COUT * CIN);
    __builtin_prefetch(wr0 + 16 * COUT * CIN, 0, 1);

    v2f a_lo, a_hi;
    a_lo.x = xr[(size_t)(kb + 0) * DBIG + Dp];
    a_lo.y = xr[(size_t)(kb + 1) * DBIG + Dp];
    a_hi.x = xr[(size_t)(kb + 4) * DBIG + Dp];
    a_hi.y = xr[(size_t)(kb + 5) * DBIG + Dp];
    v8f u0 = {};
    u0 = u_tile_wmma(a_lo, a_hi, wr0, kb, u0);

    a_lo.x = xr[(size_t)(kb + 0) * DBIG + Dp + 1];
    a_lo.y = xr[(size_t)(kb + 1) * DBIG + Dp + 1];
    a_hi.x = xr[(size_t)(kb + 4) * DBIG + Dp + 1];
    a_hi.y = xr[(size_t)(kb + 5) * DBIG + Dp + 1];
    v8f u1 = {};
    u1 = u_tile_wmma(a_lo, a_hi, wr1, kb, u1);

    // transpose both tiles through wave-private LDS (in-order within wave)
#pragma unroll
    for (int r = 0; r < 8; ++r) {
      tb[r * 32 + lane]       = u0[r];
      tb[256 + r * 32 + lane] = u1[r];
    }

    // lane's output: row b0+m of tile 'hi'  ->  dot with v over c
    float dot = 0.0f;
    const v4f* up = (const v4f*)tread;
#pragma unroll
    for (int q = 0; q < 4; ++q) {
      v4f uu = up[q];
      dot = fmaf(uu[0], vq[q][0], dot);
      dot = fmaf(uu[1], vq[q][1], dot);
      dot = fmaf(uu[2], vq[q][2], dot);
      dot = fmaf(uu[3], vq[q][3], dot);
    }
    blrow[Dp + hi] += dot;   // exclusive owner of bl[b0+m][d][Dp+hi]
  }
}

// out[b][c][d] = s[b][d][c]
__global__ void caps_out_kernel(const float* __restrict__ s,
                                float* __restrict__ out) {
  int i = blockIdx.x * blockDim.x + threadIdx.x;
  if (i >= BB * COUT * DOUT) return;
  int d = i % DOUT;
  int c = (i / DOUT) % COUT;
  int b = i / (DOUT * COUT);
  out[i] = s[((size_t)b * DOUT + d) * COUT + c];
}

// ---------------------------------------------------------------------------
extern "C" void kernel_launch(void* const* d_in, const int* in_sizes, int n_in,
                              void* d_out, int out_size, void* d_ws,
                              size_t ws_size, hipStream_t stream) {
  const float* x = (const float*)d_in[0];      // (64, 8, 1152)
  const float* w = (const float*)d_in[1];      // (64, 1152, 16, 8)
  float* ws  = (float*)d_ws;
  float* bl  = ws + OFF_BL;
  float* lse = ws + OFF_LSE;
  float* s   = ws + OFF_S;
  float* v   = ws + OFF_V;

  {
    size_t nB = (size_t)BB * DOUT * DBIG;
    caps_zero_kernel<<<(unsigned)((nB + 255) / 256), 256, 0, stream>>>(bl, nB);
  }

  dim3 grid(DOUT, BB / 16);   // (64 d) x (4 batch tiles) = 256 WGs
  for (int it = 0; it < 3; ++it) {
    caps_lse_kernel<<<(BB * DBIG + 255) / 256, 256, 0, stream>>>(bl, lse);
    caps_s_kernel<<<grid, 256, 0, stream>>>(x, w, bl, lse, s, v);
    if (it < 2) caps_b_kernel<<<grid, 256, 0, stream>>>(x, w, v, bl);
  }

  caps_out_kernel<<<(BB * COUT * DOUT + 255) / 256, 256, 0, stream>>>(
      s, (float*)d_out);
}